// FCOSPostprocessor_64226940944782
// MI455X (gfx1250) — compile-verified
//
#include <hip/hip_runtime.h>
#include <hip/hip_bf16.h>
#include <stdint.h>

#define NLOC   524288      // H*W*D = 128*128*32
#define NB     4
#define NCLS   8
#define TOPN   512
#define CAP    4096
#define NBINS  8192        // 13-bit ordered-float bins
#define SEL_K  512
#define POSTN  100
#define NMS_THR 0.6f

typedef __attribute__((ext_vector_type(2))) float v2f;
typedef __attribute__((ext_vector_type(8))) float v8f;

__device__ __forceinline__ float sigm(float x) { return 1.0f / (1.0f + __expf(-x)); }

// ---------------------------------------------------------------------------
// K1: streaming score pass -> 13-bit histogram per image (LDS-privatized)
// ---------------------------------------------------------------------------
__global__ __launch_bounds__(256) void k1_hist(const float* __restrict__ cls,
                                               const float* __restrict__ ctr,
                                               unsigned* __restrict__ hist) {
  __shared__ unsigned lh[NBINS];
  const int b = blockIdx.y;
  for (int i = threadIdx.x; i < NBINS; i += 256) lh[i] = 0u;
  __syncthreads();
  const float* clsb = cls + (size_t)b * NCLS * NLOC;
  const float* ctrb = ctr + (size_t)b * NLOC;
  const int stride = gridDim.x * 256;
  for (int n = blockIdx.x * 256 + threadIdx.x; n < NLOC; n += stride) {
    float c = sigm(ctrb[n]);
    __builtin_prefetch(&ctrb[n + stride], 0, 1);           // global_prefetch_b8
    #pragma unroll
    for (int cc = 0; cc < NCLS; ++cc) {
      float s = sigm(clsb[(size_t)cc * NLOC + n]);
      if (s > 0.05f) {
        float v = s * c;                                   // v > 0
        unsigned bits = __float_as_uint(v);
        atomicAdd(&lh[bits >> 18], 1u);
      }
    }
  }
  __syncthreads();
  unsigned* gh = hist + (size_t)b * NBINS;
  for (int i = threadIdx.x; i < NBINS; i += 256)
    if (lh[i]) atomicAdd(&gh[i], lh[i]);
}

// ---------------------------------------------------------------------------
// K2: suffix-sum of histogram via WMMA (S = U_upper * X, K-split into 4),
//     then pick threshold bin T = max bin with count_ge(bin) >= SEL_K.
//     One wave (32 lanes) per image; EXEC all-ones around WMMA.
// ---------------------------------------------------------------------------
__global__ __launch_bounds__(32) void k2_scan(const unsigned* __restrict__ hist,
                                              unsigned* __restrict__ thresh) {
  __shared__ float hf[NBINS];       // counts, overwritten in place by suffix sums
  __shared__ float gafter[NBINS / 16];
  __shared__ int   best[32];
  const int b = blockIdx.x;
  const int lane = threadIdx.x;
  const unsigned* gh = hist + (size_t)b * NBINS;
  for (int i = lane; i < NBINS; i += 32) hf[i] = (float)gh[i];
  __syncthreads();

  const int Ml    = lane & 15;            // A: M row / B: N col (doc layouts coincide)
  const int khalf = (lane < 16) ? 0 : 2;  // K element held in VGPR0 (VGPR1 = +1)

  for (int t = 0; t < NBINS / 256; ++t) { // 32 super-tiles of 16 groups x 16 elems
    const int base = t * 256;
    v8f d = {};
    #pragma unroll
    for (int s = 0; s < 4; ++s) {
      v2f a, bb;
      a.x = ((4 * s + khalf)     >= Ml) ? 1.0f : 0.0f;   // U[m][k] = k>=m, slice s
      a.y = ((4 * s + khalf + 1) >= Ml) ? 1.0f : 0.0f;
      bb.x = hf[base + Ml * 16 + 4 * s + khalf];          // X[k][g], g = lane&15
      bb.y = hf[base + Ml * 16 + 4 * s + khalf + 1];
      d = __builtin_amdgcn_wmma_f32_16x16x4_f32(false, a, false, bb,
                                                (short)0, d, false, false);
    }
    // D[M][N]: VGPR r, lanes 0-15 -> M=r, lanes 16-31 -> M=r+8; N = lane&15.
    const int Mb = (lane < 16) ? 0 : 8;
    const int N  = lane & 15;
    #pragma unroll
    for (int r = 0; r < 8; ++r)
      hf[base + N * 16 + (Mb + r)] = d[r];  // in-place: wave reads done above
  }
  __syncthreads();
  if (lane == 0) {                 // exclusive suffix over 512 group totals
    float acc = 0.0f;
    for (int g = NBINS / 16 - 1; g >= 0; --g) { gafter[g] = acc; acc += hf[g * 16]; }
  }
  __syncthreads();
  int loc = -1;
  for (int i = lane; i < NBINS; i += 32) {
    float cge = hf[i] + gafter[i >> 4];
    if (cge >= (float)SEL_K && i > loc) loc = i;
  }
  best[lane] = loc;
  __syncthreads();
  if (lane == 0) {
    int m = -1;
    for (int i = 0; i < 32; ++i) if (best[i] > m) m = best[i];
    thresh[b] = (m < 0) ? 0u : (unsigned)m;
  }
}

// ---------------------------------------------------------------------------
// K3: recompute + compact all candidates with bin >= T[b]
// ---------------------------------------------------------------------------
__global__ __launch_bounds__(256) void k3_compact(const float* __restrict__ cls,
                                                  const float* __restrict__ ctr,
                                                  const unsigned* __restrict__ thresh,
                                                  unsigned* __restrict__ cnt,
                                                  float* __restrict__ candS,
                                                  int* __restrict__ candI) {
  const int b = blockIdx.y;
  const unsigned T = thresh[b];
  const float* clsb = cls + (size_t)b * NCLS * NLOC;
  const float* ctrb = ctr + (size_t)b * NLOC;
  const int stride = gridDim.x * 256;
  for (int n = blockIdx.x * 256 + threadIdx.x; n < NLOC; n += stride) {
    float c = sigm(ctrb[n]);
    #pragma unroll
    for (int cc = 0; cc < NCLS; ++cc) {
      float s = sigm(clsb[(size_t)cc * NLOC + n]);
      if (s > 0.05f) {
        float v = s * c;
        if ((__float_as_uint(v) >> 18) >= T) {
          unsigned slot = atomicAdd(&cnt[b], 1u);
          if (slot < CAP) {
            candS[(size_t)b * CAP + slot] = v;
            candI[(size_t)b * CAP + slot] = n * NCLS + cc;
          }
        }
      }
    }
  }
}

// ---------------------------------------------------------------------------
// K4: per-image bitonic sort (desc) of up to CAP candidates; decode top 512:
//     boxes, clip, sqrt score, class, valid. Zero-init nmsKeep.
// ---------------------------------------------------------------------------
__global__ __launch_bounds__(256) void k4_select(const float* __restrict__ candS,
                                                 const int* __restrict__ candI,
                                                 const unsigned* __restrict__ cnt,
                                                 const float* __restrict__ boxp,
                                                 const float* __restrict__ locp,
                                                 float* __restrict__ det,
                                                 float* __restrict__ sc,
                                                 int* __restrict__ clsA,
                                                 int* __restrict__ valA,
                                                 int* __restrict__ nmsKeep) {
  __shared__ float ss[CAP];
  __shared__ int   si[CAP];
  const int b = blockIdx.x;
  const int tid = threadIdx.x;
  unsigned nc = cnt[b]; if (nc > CAP) nc = CAP;
  for (int k = tid; k < CAP; k += 256) {
    ss[k] = (k < (int)nc) ? candS[(size_t)b * CAP + k] : -1.0f;
    si[k] = (k < (int)nc) ? candI[(size_t)b * CAP + k] : 0;
  }
  for (int ksz = 2; ksz <= CAP; ksz <<= 1)
    for (int j = ksz >> 1; j > 0; j >>= 1) {
      __syncthreads();
      for (int i = tid; i < CAP; i += 256) {
        int p = i ^ j;
        if (p > i) {
          bool asc = ((i & ksz) != 0);            // final merge => descending
          if ((ss[i] > ss[p]) == asc) {
            float tf = ss[i]; ss[i] = ss[p]; ss[p] = tf;
            int ti = si[i]; si[i] = si[p]; si[p] = ti;
          }
        }
      }
    }
  __syncthreads();
  const float full[3] = {255.0f, 255.0f, 63.0f};
  for (int r = tid; r < TOPN; r += 256) {
    float v = ss[r];
    size_t o = (size_t)b * TOPN + r;
    nmsKeep[o] = 0;
    if (v > 0.0f) {
      int idx = si[r];
      int n = idx >> 3, cc = (idx & 7) + 1;
      #pragma unroll
      for (int dch = 0; dch < 3; ++dch) {
        float L  = locp[(size_t)n * 3 + dch];
        float b0 = boxp[(size_t)b * 6 * NLOC + (size_t)dch * NLOC + n];
        float b1 = boxp[(size_t)b * 6 * NLOC + (size_t)(dch + 3) * NLOC + n];
        float lo = fminf(fmaxf(L - b0, 0.0f), full[dch]);
        float hi = fminf(fmaxf(L + b1, 0.0f), full[dch]);
        det[o * 6 + dch]     = lo;
        det[o * 6 + dch + 3] = hi;
      }
      sc[o] = sqrtf(v);
      clsA[o] = cc;
      valA[o] = 1;
    } else {
      #pragma unroll
      for (int dch = 0; dch < 6; ++dch) det[o * 6 + dch] = 0.0f;
      sc[o] = 0.0f; clsA[o] = 0; valA[o] = 0;
    }
  }
}

// ---------------------------------------------------------------------------
// K5: per (image,class) sequential NMS on the globally sorted 512 entries.
//     Box tile staged LDS via CDNA5 async-to-LDS; s_wait_asynccnt fence.
// ---------------------------------------------------------------------------
__global__ __launch_bounds__(512) void k5_nms(const float* __restrict__ det,
                                              const float* __restrict__ sc,
                                              const int* __restrict__ clsA,
                                              const int* __restrict__ valA,
                                              int* __restrict__ nmsKeep) {
  __shared__ float det_s[TOPN][6];
  __shared__ int   cls_s[TOPN];
  __shared__ int   keep_s[TOPN];
  const int myc = blockIdx.x + 1;
  const int b   = blockIdx.y;
  const int tid = threadIdx.x;
  const float* detg = det + (size_t)b * TOPN * 6;

  // Async copy 512x6 floats global -> LDS (GV mode), then wait on ASYNCcnt.
  for (int k = tid; k < TOPN * 6; k += 512) {
    unsigned lofs = (unsigned)(uintptr_t)(&det_s[0][0] + k);
    unsigned long long ga = (unsigned long long)(uintptr_t)(detg + k);
    asm volatile("global_load_async_to_lds_b32 %0, %1, off"
                 :: "v"(lofs), "v"(ga) : "memory");
  }
  asm volatile("s_wait_asynccnt 0x0" ::: "memory");

  {
    size_t o = (size_t)b * TOPN + tid;
    cls_s[tid]  = clsA[o];
    keep_s[tid] = (valA[o] && clsA[o] == myc && sc[o] > 0.0f) ? 1 : 0;
  }
  __syncthreads();

  for (int i = 0; i < TOPN; ++i) {
    if (keep_s[i]) {
      if (tid > i && keep_s[tid]) {
        float inter = 1.0f, vi = 1.0f, ve = 1.0f;
        #pragma unroll
        for (int dch = 0; dch < 3; ++dch) {
          float li = det_s[i][dch],   hi = det_s[i][dch + 3];
          float le = det_s[tid][dch], he = det_s[tid][dch + 3];
          inter *= fmaxf(fminf(hi, he) - fmaxf(li, le), 0.0f);
          vi    *= fmaxf(hi - li, 0.0f);
          ve    *= fmaxf(he - le, 0.0f);
        }
        float iou = inter / (vi + ve - inter + 1e-9f);
        if (iou > NMS_THR) keep_s[tid] = 0;
      }
    }
    __syncthreads();
  }
  if (cls_s[tid] == myc) nmsKeep[(size_t)b * TOPN + tid] = keep_s[tid];
}

// ---------------------------------------------------------------------------
// K6: per-image top-100 of NMS survivors -> packed outputs
// ---------------------------------------------------------------------------
__global__ __launch_bounds__(512) void k6_final(const float* __restrict__ det,
                                                const float* __restrict__ sc,
                                                const int* __restrict__ clsA,
                                                const int* __restrict__ nmsKeep,
                                                float* __restrict__ out) {
  __shared__ float ss[TOPN];
  __shared__ int   si[TOPN];
  const int b = blockIdx.x;
  const int tid = threadIdx.x;
  {
    size_t o = (size_t)b * TOPN + tid;
    ss[tid] = nmsKeep[o] ? sc[o] : -1.0f;
    si[tid] = tid;
  }
  for (int ksz = 2; ksz <= TOPN; ksz <<= 1)
    for (int j = ksz >> 1; j > 0; j >>= 1) {
      __syncthreads();
      int i = tid, p = i ^ j;
      if (p > i) {
        bool asc = ((i & ksz) != 0);
        if ((ss[i] > ss[p]) == asc) {
          float tf = ss[i]; ss[i] = ss[p]; ss[p] = tf;
          int ti = si[i]; si[i] = si[p]; si[p] = ti;
        }
      }
    }
  __syncthreads();
  float* fbox = out;                       // [4][100][6]
  float* fsco = out + NB * POSTN * 6;      // [4][100]
  float* flab = fsco + NB * POSTN;         // [4][100]
  float* fval = flab + NB * POSTN;         // [4][100]
  for (int r = tid; r < POSTN; r += 512) {
    int e = si[r];
    bool ok = ss[r] > 0.0f;
    size_t o = (size_t)b * TOPN + e;
    #pragma unroll
    for (int dch = 0; dch < 6; ++dch)
      fbox[((size_t)b * POSTN + r) * 6 + dch] = ok ? det[o * 6 + dch] : 0.0f;
    fsco[(size_t)b * POSTN + r] = ok ? sc[o] : 0.0f;
    flab[(size_t)b * POSTN + r] = ok ? (float)clsA[o] : 0.0f;
    fval[(size_t)b * POSTN + r] = ok ? 1.0f : 0.0f;
  }
}

// ---------------------------------------------------------------------------
extern "C" void kernel_launch(void* const* d_in, const int* in_sizes, int n_in,
                              void* d_out, int out_size, void* d_ws, size_t ws_size,
                              hipStream_t stream) {
  const float* locp = (const float*)d_in[0];   // (N,3)
  const float* cls  = (const float*)d_in[1];   // (B,8,H,W,D)
  const float* boxp = (const float*)d_in[2];   // (B,6,H,W,D)
  const float* ctr  = (const float*)d_in[3];   // (B,1,H,W,D)

  char* ws = (char*)d_ws;
  unsigned* hist   = (unsigned*)ws;                          ws += (size_t)NB * NBINS * 4;
  unsigned* cnt    = (unsigned*)ws;                          ws += NB * 4;
  unsigned* thresh = (unsigned*)ws;                          ws += NB * 4;
  float*    candS  = (float*)ws;                             ws += (size_t)NB * CAP * 4;
  int*      candI  = (int*)ws;                               ws += (size_t)NB * CAP * 4;
  float*    det    = (float*)ws;                             ws += (size_t)NB * TOPN * 6 * 4;
  float*    sc     = (float*)ws;                             ws += (size_t)NB * TOPN * 4;
  int*      clsA   = (int*)ws;                               ws += (size_t)NB * TOPN * 4;
  int*      valA   = (int*)ws;                               ws += (size_t)NB * TOPN * 4;
  int*      nmsK   = (int*)ws;                               ws += (size_t)NB * TOPN * 4;

  hipMemsetAsync(hist, 0, (size_t)NB * NBINS * 4 + NB * 4, stream); // hist + cnt

  dim3 g1(256, NB);
  k1_hist   <<<g1, 256, 0, stream>>>(cls, ctr, hist);
  k2_scan   <<<NB, 32, 0, stream>>>(hist, thresh);
  k3_compact<<<g1, 256, 0, stream>>>(cls, ctr, thresh, cnt, candS, candI);
  k4_select <<<NB, 256, 0, stream>>>(candS, candI, cnt, boxp, locp,
                                     det, sc, clsA, valA, nmsK);
  dim3 g5(NCLS, NB);
  k5_nms    <<<g5, 512, 0, stream>>>(det, sc, clsA, valA, nmsK);
  k6_final  <<<NB, 512, 0, stream>>>(det, sc, clsA, nmsK, (float*)d_out);
}